// AttnDecoderRNN_2224793059979
// MI455X (gfx1250) — compile-verified
//
#include <hip/hip_runtime.h>
#include <hip/hip_bf16.h>
#include <math.h>

// Problem constants (match reference)
#define Hs 512
#define Vv 32000
#define Bb 64
#define Ss 32
#define Tt 32

typedef __attribute__((ext_vector_type(2))) float v2f;
typedef __attribute__((ext_vector_type(8))) float v8f;

// ---------------------------------------------------------------------------
// Generic fp32 WMMA GEMM:  C[M,N] = A[M,K] * W[N,K]^T (+ bias[n])
// One wave computes one 16x16 tile via V_WMMA_F32_16X16X4_F32 over K in steps
// of 4. Block = 256 threads = 8 waves = 16x128 tile of C.
// grid.x = M/16, grid.y = N/128. All dims are exact multiples (checked host-side).
// permT != 0 applies the (t*64+b) -> (b*permT + t) output-row permutation used
// to write logits straight into the [B,T,V] layout of d_out.
// ---------------------------------------------------------------------------
__global__ __launch_bounds__(256)
void gemm_wt_f32(const float* __restrict__ A, int lda,
                 const float* __restrict__ W, int ldw,
                 const float* __restrict__ bias,
                 float* __restrict__ C, int ldc,
                 int K, int permT) {
  const int lane  = threadIdx.x & 31;
  const int wave  = threadIdx.x >> 5;
  const int row0  = blockIdx.x * 16;                 // M tile
  const int col0  = (blockIdx.y * 8 + wave) * 16;    // N tile
  const int mrow  = row0 + (lane & 15);
  const int ncol  = col0 + (lane & 15);
  const int khalf = (lane >> 4) * 2;                 // 0 or 2

  const float* __restrict__ Ap = A + (size_t)mrow * lda + khalf;
  const float* __restrict__ Wp = W + (size_t)ncol * ldw + khalf;

  v8f acc = {};
  #pragma unroll 4
  for (int k = 0; k < K; k += 4) {
    v2f a = *(const v2f*)(Ap + k);   // A[mrow][k+khalf .. +1]
    v2f b = *(const v2f*)(Wp + k);   // B[k+khalf][ncol] striped per ISA layout
    acc = __builtin_amdgcn_wmma_f32_16x16x4_f32(
        /*neg_a=*/false, a, /*neg_b=*/false, b,
        /*c_mod=*/(short)0, acc, /*reuse_a=*/false, /*reuse_b=*/false);
  }

  const float bv = bias ? bias[ncol] : 0.0f;
  #pragma unroll
  for (int r = 0; r < 8; ++r) {
    int mr = row0 + r + ((lane >> 4) << 3);
    int orow = permT ? ((mr & 63) * permT + (mr >> 6)) : mr;   // (t,b)->(b,t)
    C[(size_t)orow * ldc + ncol] = acc[r] + bv;
  }
}

// ---------------------------------------------------------------------------
// Attention step: one workgroup per batch element b.
// scores[s] = Va . tanh(q[b]+ba+bu + keys_proj[b,s])  (+bv) ; softmax over S
// ctx = sum_s w[s] * enc[b,s,:] ; builds xch[b] = [emb[tok] | ctx | h]
// writes attention weights into d_out.
// ---------------------------------------------------------------------------
__global__ __launch_bounds__(256)
void attn_step_kernel(const float* __restrict__ enc,
                      const float* __restrict__ keys_proj,
                      const float* __restrict__ qbuf,
                      const float* __restrict__ Va,
                      const float* __restrict__ ba,
                      const float* __restrict__ bu,
                      const float* __restrict__ bvp,
                      const float* __restrict__ emb,
                      const int*   __restrict__ target,
                      const float* __restrict__ hcur,
                      float* __restrict__ xch,
                      float* __restrict__ attn_out,
                      int t) {
  const int b   = blockIdx.x;
  const int tid = threadIdx.x;

  __shared__ float sq[Hs];   // q + ba + bu
  __shared__ float sw[Ss];   // scores -> softmax weights

  for (int h = tid; h < Hs; h += 256) sq[h] = qbuf[b * Hs + h] + ba[h] + bu[h];
  __syncthreads();

  // 8 threads per source position s
  const int s   = tid >> 3;   // 0..31
  const int sub = tid & 7;    // 0..7
  {
    const float* kp = keys_proj + ((size_t)b * Ss + s) * Hs;
    float p = 0.0f;
    for (int h = sub; h < Hs; h += 8) p += Va[h] * tanhf(sq[h] + kp[h]);
    p += __shfl_down(p, 4, 8);
    p += __shfl_down(p, 2, 8);
    p += __shfl_down(p, 1, 8);
    if (sub == 0) sw[s] = p + bvp[0];
  }
  __syncthreads();

  // softmax over S=32 within one wave
  if (tid < 32) {
    float v = sw[tid];
    float m = v;
    for (int off = 16; off > 0; off >>= 1) m = fmaxf(m, __shfl_xor(m, off, 32));
    float e = __expf(v - m);
    float sum = e;
    for (int off = 16; off > 0; off >>= 1) sum += __shfl_xor(sum, off, 32);
    float w = e / sum;
    sw[tid] = w;
    attn_out[(size_t)b * (Tt * Ss) + t * Ss + tid] = w;   // attn[b][t][s]
  }
  __syncthreads();

  const int tok = (t == 0) ? 0 : target[b * Tt + (t - 1)];
  for (int h = tid; h < Hs; h += 256) {
    const float* eb = enc + (size_t)b * (Ss * Hs) + h;
    float c = 0.0f;
    #pragma unroll
    for (int s2 = 0; s2 < Ss; ++s2) c += sw[s2] * eb[s2 * Hs];
    float* xr = xch + (size_t)b * (3 * Hs);
    xr[h]            = emb[(size_t)tok * Hs + h];  // x
    xr[Hs + h]       = c;                           // ctx
    xr[2 * Hs + h]   = hcur[b * Hs + h];            // h (for gh GEMM reuse)
  }
}

// ---------------------------------------------------------------------------
// GRU gate elementwise: h_new = (1-z)*n + z*h
// ---------------------------------------------------------------------------
__global__ __launch_bounds__(256)
void gru_elem_kernel(const float* __restrict__ gi,
                     const float* __restrict__ gh,
                     const float* __restrict__ b_ih,
                     const float* __restrict__ b_hh,
                     const float* __restrict__ hprev,
                     float* __restrict__ hnew) {
  const int idx = blockIdx.x * 256 + threadIdx.x;    // 0..B*H-1
  const int b = idx >> 9, j = idx & (Hs - 1);
  const size_t r3 = (size_t)b * (3 * Hs);
  float ir = gi[r3 + j]          + b_ih[j];
  float iz = gi[r3 + Hs + j]     + b_ih[Hs + j];
  float in_= gi[r3 + 2*Hs + j]   + b_ih[2*Hs + j];
  float hr = gh[r3 + j]          + b_hh[j];
  float hz = gh[r3 + Hs + j]     + b_hh[Hs + j];
  float hn = gh[r3 + 2*Hs + j]   + b_hh[2*Hs + j];
  float r = 1.0f / (1.0f + __expf(-(ir + hr)));
  float z = 1.0f / (1.0f + __expf(-(iz + hz)));
  float n = tanhf(in_ + r * hn);
  float h = hprev[b * Hs + j];
  hnew[b * Hs + j] = (1.0f - z) * n + z * h;
}

// ---------------------------------------------------------------------------
// In-place log_softmax over V=32000 per output row of d_out (rows are b*T+t).
// ---------------------------------------------------------------------------
__global__ __launch_bounds__(256)
void log_softmax_kernel(float* __restrict__ out) {
  float* p = out + (size_t)blockIdx.x * Vv;
  const int tid = threadIdx.x, lane = tid & 31, wave = tid >> 5;
  __shared__ float red[8];

  float m = -INFINITY;
  for (int i = tid; i < Vv; i += 256) m = fmaxf(m, p[i]);
  for (int off = 16; off > 0; off >>= 1) m = fmaxf(m, __shfl_xor(m, off, 32));
  if (lane == 0) red[wave] = m;
  __syncthreads();
  m = red[0];
  #pragma unroll
  for (int w = 1; w < 8; ++w) m = fmaxf(m, red[w]);
  __syncthreads();

  float s = 0.0f;
  for (int i = tid; i < Vv; i += 256) s += __expf(p[i] - m);
  for (int off = 16; off > 0; off >>= 1) s += __shfl_xor(s, off, 32);
  if (lane == 0) red[wave] = s;
  __syncthreads();
  s = red[0];
  #pragma unroll
  for (int w = 1; w < 8; ++w) s += red[w];

  const float shift = m + logf(s);
  for (int i = tid; i < Vv; i += 256) p[i] = p[i] - shift;
}

// ---------------------------------------------------------------------------
extern "C" void kernel_launch(void* const* d_in, const int* in_sizes, int n_in,
                              void* d_out, int out_size, void* d_ws, size_t ws_size,
                              hipStream_t stream) {
  (void)in_sizes; (void)n_in; (void)out_size; (void)ws_size;

  const float* enc   = (const float*)d_in[0];   // [B,S,H]
  const float* eh    = (const float*)d_in[1];   // [1,B,H]
  const int*   tgt   = (const int*)  d_in[2];   // [B,T]
  const float* emb   = (const float*)d_in[3];   // [V,H]
  const float* Wa    = (const float*)d_in[4];   // [H,H]
  const float* ba    = (const float*)d_in[5];
  const float* Ua    = (const float*)d_in[6];   // [H,H]
  const float* bu    = (const float*)d_in[7];
  const float* Va    = (const float*)d_in[8];   // [1,H]
  const float* bvp   = (const float*)d_in[9];
  const float* W_ih  = (const float*)d_in[10];  // [3H,2H]
  const float* W_hh  = (const float*)d_in[11];  // [3H,H]
  const float* b_ih  = (const float*)d_in[12];
  const float* b_hh  = (const float*)d_in[13];
  const float* W_out = (const float*)d_in[14];  // [V,H]
  const float* b_out = (const float*)d_in[15];

  float* out      = (float*)d_out;
  float* hT_out   = out + (size_t)Bb * Tt * Vv;                 // [1,B,H]
  float* attn_out = hT_out + (size_t)Bb * Hs;                   // [B,T,S]

  // scratch layout (floats)
  float* ws     = (float*)d_ws;
  float* keys   = ws;                               // B*S*H   = 1,048,576
  float* Hchain = keys   + (size_t)Bb * Ss * Hs;    // (T+1)*B*H = 1,081,344
  float* qbuf   = Hchain + (size_t)(Tt + 1) * Bb * Hs;  // B*H
  float* xch    = qbuf   + (size_t)Bb * Hs;         // B*3H
  float* gibuf  = xch    + (size_t)Bb * 3 * Hs;     // B*3H
  float* ghbuf  = gibuf  + (size_t)Bb * 3 * Hs;     // B*3H

  const dim3 blk(256);

  // h0 = encoder_hidden[0]
  hipMemcpyAsync(Hchain, eh, (size_t)Bb * Hs * sizeof(float),
                 hipMemcpyDeviceToDevice, stream);

  // keys_proj = enc @ Ua^T   (bias bu folded into attention kernel)
  gemm_wt_f32<<<dim3((Bb * Ss) / 16, Hs / 128), blk, 0, stream>>>(
      enc, Hs, Ua, Hs, nullptr, keys, Hs, Hs, 0);

  for (int t = 0; t < Tt; ++t) {
    const float* hcur = Hchain + (size_t)t * Bb * Hs;
    float*       hnew = Hchain + (size_t)(t + 1) * Bb * Hs;

    // q = h @ Wa^T   (bias ba folded into attention kernel)
    gemm_wt_f32<<<dim3(Bb / 16, Hs / 128), blk, 0, stream>>>(
        hcur, Hs, Wa, Hs, nullptr, qbuf, Hs, Hs, 0);

    // attention + softmax + ctx + build xch = [x | ctx | h]
    attn_step_kernel<<<dim3(Bb), blk, 0, stream>>>(
        enc, keys, qbuf, Va, ba, bu, bvp, emb, tgt, hcur, xch, attn_out, t);

    // gi = [x|ctx] @ W_ih^T ; gh = h @ W_hh^T  (biases in elementwise kernel)
    gemm_wt_f32<<<dim3(Bb / 16, (3 * Hs) / 128), blk, 0, stream>>>(
        xch, 3 * Hs, W_ih, 2 * Hs, nullptr, gibuf, 3 * Hs, 2 * Hs, 0);
    gemm_wt_f32<<<dim3(Bb / 16, (3 * Hs) / 128), blk, 0, stream>>>(
        xch + 2 * Hs, 3 * Hs, W_hh, Hs, nullptr, ghbuf, 3 * Hs, Hs, 0);

    // GRU gates -> h_new
    gru_elem_kernel<<<dim3((Bb * Hs) / 256), blk, 0, stream>>>(
        gibuf, ghbuf, b_ih, b_hh, hcur, hnew);
  }

  // batched logits for all T: [T*B, H] x W_out^T -> d_out with (t,b)->(b,t)
  gemm_wt_f32<<<dim3((Tt * Bb) / 16, Vv / 128), blk, 0, stream>>>(
      Hchain + (size_t)Bb * Hs, Hs, W_out, Hs, b_out, out, Vv, Hs, /*permT=*/Tt);

  // in-place log_softmax over V per (b,t) row
  log_softmax_kernel<<<dim3(Bb * Tt), blk, 0, stream>>>(out);

  // hT = Hchain[T]
  hipMemcpyAsync(hT_out, Hchain + (size_t)Tt * Bb * Hs,
                 (size_t)Bb * Hs * sizeof(float),
                 hipMemcpyDeviceToDevice, stream);
}